// LocalGraphProjection_28750511079686
// MI455X (gfx1250) — compile-verified
//
#include <hip/hip_runtime.h>
#include <math.h>

#define N_PTS 262144
#define PTS_PER_BLOCK 16
#define THREADS 256

typedef float v2f __attribute__((ext_vector_type(2)));
typedef float v8f __attribute__((ext_vector_type(8)));

// ---------------------------------------------------------------------------
// Setup kernel: build, per view v, a 4x3 combined homogeneous transform
//   pc_v = [x,y,z,1] @ Mv  where Mv = [ inv(c0^T) @ Rv^T ; (o0 - ov) @ Rv^T ]
// stored in ws as cm[(v*4 + k)*3 + c]  (k = row/K-index, c = output component)
// ---------------------------------------------------------------------------
__global__ void lgp_setup_kernel(const float* __restrict__ cams,
                                 float* __restrict__ cm) {
    if (threadIdx.x != 0 || blockIdx.x != 0) return;
    const float PI_F = (float)M_PI;
    const float D2R  = (float)(M_PI / 180.0);

    float R[3][3][3];  // R[v][row][col]
    float O[3][3];     // O[v][comp] (unnormalized Zv = camera origin)

    for (int v = 0; v < 3; ++v) {
        float az = cams[v * 5 + 0];
        float el = cams[v * 5 + 1];
        float dist = cams[v * 5 + 3];
        float theta = az * D2R;
        float camy = dist * sinf(el * D2R);
        float lens = dist * cosf(el * D2R);
        float camx = lens * cosf(theta);
        float camz = lens * sinf(theta);
        float Zv[3] = {camx, camy, camz};
        float Yv[3] = {camy * cosf(theta + PI_F), lens, camy * sinf(theta + PI_F)};
        float Xv[3] = {Yv[1] * Zv[2] - Yv[2] * Zv[1],
                       Yv[2] * Zv[0] - Yv[0] * Zv[2],
                       Yv[0] * Zv[1] - Yv[1] * Zv[0]};
        float nx = sqrtf(Xv[0]*Xv[0] + Xv[1]*Xv[1] + Xv[2]*Xv[2]);
        float ny = sqrtf(Yv[0]*Yv[0] + Yv[1]*Yv[1] + Yv[2]*Yv[2]);
        float nz = sqrtf(Zv[0]*Zv[0] + Zv[1]*Zv[1] + Zv[2]*Zv[2]);
        for (int c = 0; c < 3; ++c) {
            R[v][0][c] = Xv[c] / nx;
            R[v][1][c] = Yv[c] / ny;
            R[v][2][c] = Zv[c] / nz;
            O[v][c] = Zv[c];
        }
    }

    // A0 = inv(M), M = c0^T  (M[i][j] = R[0][j][i])
    float M[3][3];
    for (int i = 0; i < 3; ++i)
        for (int j = 0; j < 3; ++j) M[i][j] = R[0][j][i];
    float det = M[0][0] * (M[1][1]*M[2][2] - M[1][2]*M[2][1])
              - M[0][1] * (M[1][0]*M[2][2] - M[1][2]*M[2][0])
              + M[0][2] * (M[1][0]*M[2][1] - M[1][1]*M[2][0]);
    float A0[3][3];
    A0[0][0] = (M[1][1]*M[2][2] - M[1][2]*M[2][1]) / det;
    A0[0][1] = (M[0][2]*M[2][1] - M[0][1]*M[2][2]) / det;
    A0[0][2] = (M[0][1]*M[1][2] - M[0][2]*M[1][1]) / det;
    A0[1][0] = (M[1][2]*M[2][0] - M[1][0]*M[2][2]) / det;
    A0[1][1] = (M[0][0]*M[2][2] - M[0][2]*M[2][0]) / det;
    A0[1][2] = (M[0][2]*M[1][0] - M[0][0]*M[1][2]) / det;
    A0[2][0] = (M[1][0]*M[2][1] - M[1][1]*M[2][0]) / det;
    A0[2][1] = (M[0][1]*M[2][0] - M[0][0]*M[2][1]) / det;
    A0[2][2] = (M[0][0]*M[1][1] - M[0][1]*M[1][0]) / det;

    for (int v = 0; v < 3; ++v) {
        for (int c = 0; c < 3; ++c) {
            // rows 0..2: inv(c0^T) @ Rv^T
            for (int j = 0; j < 3; ++j) {
                float s = 0.0f;
                for (int m = 0; m < 3; ++m) s += A0[j][m] * R[v][c][m];
                cm[(v * 4 + j) * 3 + c] = s;
            }
            // row 3: translation (o0 - ov) @ Rv^T
            float t = 0.0f;
            for (int m = 0; m < 3; ++m) t += (O[0][m] - O[v][m]) * R[v][c][m];
            cm[(v * 4 + 3) * 3 + c] = t;
        }
    }
}

// ---------------------------------------------------------------------------
// Main kernel: 1 block = 16 points.
//   Phase 0 (wave 0): WMMA 16x16x4 homogeneous transform -> LDS pc[16][9]
//   Phase 1 (144 thr): per (point,view,level) bilinear weights + offsets
//   Phase 2 (all): 448 channel-chunk gathers + 48 coord copies, NT stores
// ---------------------------------------------------------------------------
__global__ __launch_bounds__(THREADS) void lgp_proj_kernel(
    const float* __restrict__ coord,
    const float* __restrict__ f0,
    const float* __restrict__ f1,
    const float* __restrict__ f2,
    const float* __restrict__ cm,
    float* __restrict__ out) {

    __shared__ float lds_pc[PTS_PER_BLOCK * 12];      // pc[point][3v+c], padded to 12
    __shared__ float smp_w[PTS_PER_BLOCK * 9 * 4];    // weights per (pt, v*3+l)
    __shared__ int   smp_o[PTS_PER_BLOCK * 9 * 4];    // element offsets per corner

    const int tid = threadIdx.x;
    const int p0 = blockIdx.x * PTS_PER_BLOCK;

    // -------- Phase 0: WMMA homogeneous transform (wave 0, EXEC all ones) ---
    if (tid < 32) {
        const int lane = tid;
        const int half = lane >> 4;
        const int m = lane & 15;
        const int p = p0 + m;

        // A (16x4) f32 layout: lanes 0-15 -> K={0,1}, lanes 16-31 -> K={2,3}
        v2f a;
        a.x = (half == 0) ? coord[p * 3 + 0] : coord[p * 3 + 2];
        a.y = (half == 0) ? coord[p * 3 + 1] : 1.0f;

        // B (4x16) f32 layout: N = lane%16; lanes 0-15 -> K={0,1}, 16-31 -> K={2,3}
        const int n = m;
        v2f b;
        if (n < 9) {
            const int v = n / 3, c = n % 3;
            b.x = cm[(v * 4 + 2 * half + 0) * 3 + c];
            b.y = cm[(v * 4 + 2 * half + 1) * 3 + c];
        } else {
            b.x = 0.0f; b.y = 0.0f;
        }

        v8f cz = {0.0f, 0.0f, 0.0f, 0.0f, 0.0f, 0.0f, 0.0f, 0.0f};
        v8f d = __builtin_amdgcn_wmma_f32_16x16x4_f32(
            false, a, false, b, (short)0, cz, false, false);

        // D layout: lane L, VGPR r -> (M = r + 8*(L/16), N = L%16)
        if (n < 9) {
            #pragma unroll
            for (int r = 0; r < 8; ++r)
                lds_pc[(r + 8 * half) * 12 + n] = d[r];
        }
    }
    __syncthreads();

    // -------- Phase 1: projection + bilinear setup per (point, view, level) --
    if (tid < PTS_PER_BLOCK * 9) {
        const int i = tid / 9;
        const int r = tid % 9;          // r = v*3 + l
        const int v = r / 3;
        const int l = r % 3;

        const float X = lds_pc[i * 12 + 3 * v + 0];
        const float Y = lds_pc[i * 12 + 3 * v + 1];
        const float Z = lds_pc[i * 12 + 3 * v + 2];
        const float negZ = -Z;

        float hh = 248.0f * ((-Y) / negZ) + 112.0f;
        hh = fminf(fmaxf(hh, 0.0f), 223.0f);
        float ww = 248.0f * (X / negZ) + 112.0f;
        ww = fminf(fmaxf(ww, 0.0f), 223.0f);

        const float scale = (float)(1 << l);
        const int H = 224 >> l;
        const int C = 16 << l;

        const float x = hh / scale;
        const float y = ww / scale;
        const float x1 = floorf(x), x2 = ceilf(x);
        const float y1 = floorf(y), y2 = ceilf(y);
        int i1 = (int)x1, j1 = (int)y1;
        int i2 = (int)x2; i2 = (i2 < H - 1) ? i2 : (H - 1);
        int j2 = (int)y2; j2 = (j2 < H - 1) ? j2 : (H - 1);

        const int base = v * H * H * C;
        const int s = (i * 9 + r) * 4;
        smp_w[s + 0] = (x2 - x) * (y2 - y);   // w11 -> Q[i1][j1]
        smp_w[s + 1] = (x - x1) * (y2 - y);   // w21 -> Q[i2][j1]
        smp_w[s + 2] = (x2 - x) * (y - y1);   // w12 -> Q[i1][j2]
        smp_w[s + 3] = (x - x1) * (y - y1);   // w22 -> Q[i2][j2]
        smp_o[s + 0] = base + (i1 * H + j1) * C;
        smp_o[s + 1] = base + (i2 * H + j1) * C;
        smp_o[s + 2] = base + (i1 * H + j2) * C;
        smp_o[s + 3] = base + (i2 * H + j2) * C;
    }
    __syncthreads();

    // -------- Phase 2: gathers + stats; NT stores of the streaming output ---
    // 448 items: (point, channel-chunk-of-4); 48 items: coord passthrough.
    for (int idx = tid; idx < PTS_PER_BLOCK * 28 + PTS_PER_BLOCK * 3; idx += THREADS) {
        if (idx < PTS_PER_BLOCK * 28) {
            const int i = idx / 28;
            const int chunk = idx % 28;
            int l, c4;
            if (chunk < 4)       { l = 0; c4 = chunk; }
            else if (chunk < 12) { l = 1; c4 = chunk - 4; }
            else                 { l = 2; c4 = chunk - 12; }
            const float* fp = (l == 0) ? f0 : ((l == 1) ? f1 : f2);
            const int coff = c4 * 4;

            float acc[3][4];
            #pragma unroll
            for (int v = 0; v < 3; ++v) {
                const int s = (i * 9 + v * 3 + l) * 4;
                const float w11 = smp_w[s + 0], w21 = smp_w[s + 1];
                const float w12 = smp_w[s + 2], w22 = smp_w[s + 3];
                const float4 q11 = *(const float4*)(fp + smp_o[s + 0] + coff);
                const float4 q21 = *(const float4*)(fp + smp_o[s + 1] + coff);
                const float4 q12 = *(const float4*)(fp + smp_o[s + 2] + coff);
                const float4 q22 = *(const float4*)(fp + smp_o[s + 3] + coff);
                acc[v][0] = w11*q11.x + w21*q21.x + w12*q12.x + w22*q22.x;
                acc[v][1] = w11*q11.y + w21*q21.y + w12*q12.y + w22*q22.y;
                acc[v][2] = w11*q11.z + w21*q21.z + w12*q12.z + w22*q22.z;
                acc[v][3] = w11*q11.w + w21*q21.w + w12*q12.w + w22*q22.w;
            }

            const int och = ((l == 0) ? 0 : ((l == 1) ? 16 : 48)) + coff;
            const size_t ob = (size_t)(p0 + i) * 339;
            #pragma unroll
            for (int k = 0; k < 4; ++k) {
                const float a0 = acc[0][k], a1 = acc[1][k], a2 = acc[2][k];
                const float mx = fmaxf(fmaxf(a0, a1), a2);
                const float mn = (a0 + a1 + a2) / 3.0f;
                const float d0 = a0 - mn, d1 = a1 - mn, d2 = a2 - mn;
                const float sd = sqrtf((d0*d0 + d1*d1 + d2*d2) / 3.0f);
                __builtin_nontemporal_store(mx, out + ob + 3   + och + k);
                __builtin_nontemporal_store(mn, out + ob + 115 + och + k);
                __builtin_nontemporal_store(sd, out + ob + 227 + och + k);
            }
        } else {
            const int ci = idx - PTS_PER_BLOCK * 28;
            const int i = ci / 3, comp = ci % 3;
            const size_t p = (size_t)(p0 + i);
            __builtin_nontemporal_store(coord[p * 3 + comp], out + p * 339 + comp);
        }
    }
}

extern "C" void kernel_launch(void* const* d_in, const int* in_sizes, int n_in,
                              void* d_out, int out_size, void* d_ws, size_t ws_size,
                              hipStream_t stream) {
    (void)in_sizes; (void)n_in; (void)out_size; (void)ws_size;
    const float* coord = (const float*)d_in[0];
    const float* f0    = (const float*)d_in[1];
    const float* f1    = (const float*)d_in[2];
    const float* f2    = (const float*)d_in[3];
    const float* cams  = (const float*)d_in[4];
    float* out = (float*)d_out;
    float* cm  = (float*)d_ws;   // 36 floats: combined per-view 4x3 transforms

    lgp_setup_kernel<<<1, 1, 0, stream>>>(cams, cm);
    lgp_proj_kernel<<<N_PTS / PTS_PER_BLOCK, THREADS, 0, stream>>>(
        coord, f0, f1, f2, cm, out);
}